// ffodo_4054449128286
// MI455X (gfx1250) — compile-verified
//
#include <hip/hip_runtime.h>
#include <math.h>

typedef __attribute__((ext_vector_type(2))) float v2f;
typedef __attribute__((ext_vector_type(8))) float v8f;

#define NV 4096      // decision variables / LS rows (square)
#define KC 8         // constraints
#define MAXIT 300
#define LR_P 0.01f
#define LR_D 0.001f
#define AB1 0.9f
#define AB2 0.999f
#define AEPS 1e-8f

#define BT 128       // GEMM block tile (i and j)
#define KB 16        // GEMM k panel staged in LDS

// ---------------------------------------------------------------------------
// G = 2 * M^T M via V_WMMA_F32_16X16X4_F32.
// Block: 256 threads = 8 waves. Block tile 128x128. Wave (wi=0..3, wj=0..1)
// owns a 32x64 region => 2x4 accumulator tiles of 16x16 (8 VGPRs each).
// A-operand tile (16x4) layout: lanes 0-15 hold M=0..15 with VGPR0=K0,VGPR1=K1;
// lanes 16-31 hold K2/K3. B-operand (4x16): VGPR0 = rows K0(lanes0-15)/K1,
// VGPR1 = rows K2/K3. C/D: VGPR v -> row v (lanes0-15) / v+8 (lanes16-31).
// Staging uses float4 global loads + b128 LDS stores (rows 16B-aligned:
// stride (BT+4)*4 = 528 B, cols multiples of 4).
// ---------------------------------------------------------------------------
__global__ __launch_bounds__(256) void gram_wmma(const float* __restrict__ M,
                                                 float* __restrict__ G) {
  __shared__ float lA[KB][BT + 4];
  __shared__ float lB[KB][BT + 4];
  const int i0 = blockIdx.y * BT;
  const int j0 = blockIdx.x * BT;
  const int tid  = threadIdx.x;
  const int wave = tid >> 5;
  const int lane = tid & 31;
  const int h = lane >> 4;          // half-wave: 0 or 1
  const int m = lane & 15;
  const int wi = wave >> 1;         // 0..3 -> i offset wi*32
  const int wj = wave & 1;          // 0..1 -> j offset wj*64

  v8f acc[2][4];
#pragma unroll
  for (int si = 0; si < 2; ++si)
#pragma unroll
    for (int sj = 0; sj < 4; ++sj)
#pragma unroll
      for (int v = 0; v < 8; ++v) acc[si][sj][v] = 0.0f;

  for (int k0 = 0; k0 < NV; k0 += KB) {
    __syncthreads();
    // Stage two 16x128 panels of M rows (columns i0.. and j0..) into LDS.
    // 512 float4 quads per panel, 256 threads -> 2 quads each, b128 path.
#pragma unroll
    for (int q = tid; q < KB * BT / 4; q += 256) {
      const int r = q >> 5;              // k-row within panel
      const int c = (q & 31) * 4;        // column (multiple of 4)
      const float4 va = *(const float4*)&M[(size_t)(k0 + r) * NV + (i0 + c)];
      const float4 vb = *(const float4*)&M[(size_t)(k0 + r) * NV + (j0 + c)];
      *(float4*)&lA[r][c] = va;
      *(float4*)&lB[r][c] = vb;
    }
    __syncthreads();

#pragma unroll
    for (int kk = 0; kk < KB; kk += 4) {
      v2f a[2], b[4];
#pragma unroll
      for (int si = 0; si < 2; ++si) {
        const int iB = wi * 32 + si * 16 + m;
        a[si].x = lA[kk + 2 * h][iB];
        a[si].y = lA[kk + 2 * h + 1][iB];
      }
#pragma unroll
      for (int sj = 0; sj < 4; ++sj) {
        const int jB = wj * 64 + sj * 16 + m;
        b[sj].x = lB[kk + h][jB];
        b[sj].y = lB[kk + 2 + h][jB];
      }
#pragma unroll
      for (int si = 0; si < 2; ++si)
#pragma unroll
        for (int sj = 0; sj < 4; ++sj)
          acc[si][sj] = __builtin_amdgcn_wmma_f32_16x16x4_f32(
              false, a[si], false, b[sj], (short)0, acc[si][sj], false, false);
    }
  }

  // Store G = 2 * acc (factor 2 from d/dz of ||Mz-y||^2).
#pragma unroll
  for (int si = 0; si < 2; ++si)
#pragma unroll
    for (int sj = 0; sj < 4; ++sj)
#pragma unroll
      for (int v = 0; v < 8; ++v) {
        const int row = i0 + wi * 32 + si * 16 + v + 8 * h;
        const int col = j0 + wj * 64 + sj * 16 + m;
        G[(size_t)row * NV + col] = 2.0f * acc[si][sj][v];
      }
}

// c2[i] = 2 * sum_k M[k][i] * y[k]   (coalesced column reduction)
__global__ void mty_kernel(const float* __restrict__ M,
                           const float* __restrict__ y,
                           float* __restrict__ c2) {
  const int i = blockIdx.x * blockDim.x + threadIdx.x;
  float s = 0.0f;
  for (int k = 0; k < NV; ++k) s += M[(size_t)k * NV + i] * y[k];
  c2[i] = 2.0f * s;
}

__device__ __forceinline__ float waveReduce(float v) {
#pragma unroll
  for (int off = 16; off > 0; off >>= 1) v += __shfl_xor(v, off, 32);
  return v;
}

// One optimizer step. Blocks 0..511: one wave per row r of G computes
// gz[r] = (Gz)_r - c2[r] + (A^T mult)_r then Adam-updates z (ping-pong).
// Last block: 8 waves compute cv = A z - b and do the dual Adam + clamp.
__global__ __launch_bounds__(256) void iter_kernel(
    const float* __restrict__ G, const float* __restrict__ c2,
    const float* __restrict__ A, const float* __restrict__ bvec,
    const float* __restrict__ zin, float* __restrict__ zout,
    float* __restrict__ mz, float* __restrict__ vz,
    const float* __restrict__ min_, float* __restrict__ mout,
    float* __restrict__ mm, float* __restrict__ vm,
    float bc1, float bc2) {
  const int wave = threadIdx.x >> 5;
  const int lane = threadIdx.x & 31;
  const float4* z4 = (const float4*)zin;

  if (blockIdx.x == gridDim.x - 1) {
    // ---- dual side: wave k handles constraint k ----
    const float4* A4 = (const float4*)(A + (size_t)wave * NV);
    float s = 0.0f;
#pragma unroll 4
    for (int c = lane; c < NV / 4; c += 32) {
      const float4 av = A4[c], zv = z4[c];
      s += av.x * zv.x + av.y * zv.y + av.z * zv.z + av.w * zv.w;
    }
    s = waveReduce(s);
    if (lane == 0) {
      const float cv = s - bvec[wave];
      const float g  = -(cv - 2.0f * min_[wave]);   // ascent direction
      const float m_ = AB1 * mm[wave] + (1.0f - AB1) * g;
      const float v_ = AB2 * vm[wave] + (1.0f - AB2) * g * g;
      mm[wave] = m_;
      vm[wave] = v_;
      const float upd = min_[wave] - LR_D * (m_ * bc1) / (sqrtf(v_ * bc2) + AEPS);
      mout[wave] = fmaxf(upd, 0.0f);                // clamp_(min=0)
    }
  } else {
    // ---- primal side: one wave per row of G ----
    const int r = blockIdx.x * 8 + wave;
    const float4* G4 = (const float4*)(G + (size_t)r * NV);
    float s = 0.0f;
#pragma unroll 4
    for (int c = lane; c < NV / 4; c += 32) {
      const float4 gv = G4[c], zv = z4[c];
      s += gv.x * zv.x + gv.y * zv.y + gv.z * zv.z + gv.w * zv.w;
    }
    s = waveReduce(s);
    if (lane == 0) {
      float g = s - c2[r];
#pragma unroll
      for (int k = 0; k < KC; ++k) g += A[(size_t)k * NV + r] * min_[k];
      const float m_ = AB1 * mz[r] + (1.0f - AB1) * g;
      const float v_ = AB2 * vz[r] + (1.0f - AB2) * g * g;
      mz[r] = m_;
      vz[r] = v_;
      zout[r] = zin[r] - LR_P * (m_ * bc1) / (sqrtf(v_ * bc2) + AEPS);
    }
  }
}

__global__ void init_kernel(const float* __restrict__ z0,
                            const float* __restrict__ g0,
                            float* __restrict__ zb, float* __restrict__ mb,
                            float* __restrict__ mz, float* __restrict__ vz,
                            float* __restrict__ mm, float* __restrict__ vm) {
  const int i = blockIdx.x * blockDim.x + threadIdx.x;
  if (i < NV) { zb[i] = z0[i]; mz[i] = 0.0f; vz[i] = 0.0f; }
  if (i < KC) { mb[i] = g0[i]; mm[i] = 0.0f; vm[i] = 0.0f; }
}

__global__ void finalize_kernel(const float* __restrict__ z,
                                const float* __restrict__ mult,
                                float* __restrict__ out) {
  const int i = blockIdx.x * blockDim.x + threadIdx.x;
  if (i < NV) out[i] = z[i];
  if (i < KC) out[NV + i] = mult[i];
}

extern "C" void kernel_launch(void* const* d_in, const int* in_sizes, int n_in,
                              void* d_out, int out_size, void* d_ws, size_t ws_size,
                              hipStream_t stream) {
  const float* M  = (const float*)d_in[0];   // (4096,4096)
  const float* y  = (const float*)d_in[1];   // (4096,1)
  const float* A  = (const float*)d_in[2];   // (8,4096)
  const float* b  = (const float*)d_in[3];   // (8,)
  const float* z0 = (const float*)d_in[4];   // (4096,1)
  const float* g0 = (const float*)d_in[5];   // (8,)

  // workspace layout (floats): G | c2 | zb0 | zb1 | mz | vz | mb0 | mb1 | mm | vm
  float* G   = (float*)d_ws;
  float* c2  = G + (size_t)NV * NV;
  float* zb0 = c2 + NV;
  float* zb1 = zb0 + NV;
  float* mz  = zb1 + NV;
  float* vz  = mz + NV;
  float* mb0 = vz + NV;
  float* mb1 = mb0 + KC;
  float* mm  = mb1 + KC;
  float* vm  = mm + KC;

  init_kernel<<<16, 256, 0, stream>>>(z0, g0, zb0, mb0, mz, vz, mm, vm);
  mty_kernel<<<NV / 256, 256, 0, stream>>>(M, y, c2);
  gram_wmma<<<dim3(NV / BT, NV / BT), 256, 0, stream>>>(M, G);

  double p1 = 1.0, p2 = 1.0;
  for (int t = 1; t <= MAXIT; ++t) {
    p1 *= (double)AB1;
    p2 *= (double)AB2;
    const float bc1 = (float)(1.0 / (1.0 - p1));
    const float bc2 = (float)(1.0 / (1.0 - p2));
    const float* zin  = (t & 1) ? zb0 : zb1;
    float*       zo   = (t & 1) ? zb1 : zb0;
    const float* mi   = (t & 1) ? mb0 : mb1;
    float*       mo   = (t & 1) ? mb1 : mb0;
    iter_kernel<<<NV / 8 + 1, 256, 0, stream>>>(G, c2, A, b, zin, zo,
                                                mz, vz, mi, mo, mm, vm, bc1, bc2);
  }
  // MAXIT = 300 (even) -> final state lands in zb0 / mb0
  finalize_kernel<<<16, 256, 0, stream>>>(zb0, mb0, (float*)d_out);
}